// GaussianRasterizer_78245714199369
// MI455X (gfx1250) — compile-verified
//
#include <hip/hip_runtime.h>
#include <hip/hip_bf16.h>

typedef __attribute__((ext_vector_type(2))) float v2f;
typedef __attribute__((ext_vector_type(8))) float v8f;

#define TILE_PX 16.0f

struct GSCtx {
    // uniform scalars
    float limx, limy, focal_x, focal_y, W, H, gx, gy, scale_mod;
    // view matrix entries needed per point (row-major 4x4)
    float vm0, vm1, vm2, vm4, vm5, vm6, vm8, vm9, vm10;
};

template <bool TAIL>
__device__ __forceinline__ void gs_process32(
    int base, int N, int stride,
    int lane, int laneMod16, bool hi, int meanOff,
    v2f A, const GSCtx& C,
    const float* __restrict__ means3D,
    const float* __restrict__ scales,
    const float* __restrict__ rotations,
    float* __restrict__ radii)
{
    // ---- B matrices: 16 points per WMMA, point = column.
    // lane n (<16): (x_p, y_p); lane n+16: (z_p, 1.0). Selects only -> EXEC stays full.
    int p1 = base + laneMod16;
    int p2 = base + 16 + laneMod16;
    if (TAIL) { if (p1 > N - 1) p1 = N - 1; if (p2 > N - 1) p2 = N - 1; }

    const float* mp1 = means3D + 3 * p1;
    const float* mp2 = means3D + 3 * p2;
    float b1x = mp1[meanOff];
    float b1y = mp1[1];
    float b2x = mp2[meanOff];
    float b2y = mp2[1];
    v2f B1; B1.x = b1x; B1.y = hi ? 1.0f : b1y;
    v2f B2; B2.x = b2x; B2.y = hi ? 1.0f : b2y;

    const v8f Czero = {0.f,0.f,0.f,0.f,0.f,0.f,0.f,0.f};
    // D = A(16x4) * B(4x16): column n = point n, row m = component m
    v8f D1 = __builtin_amdgcn_wmma_f32_16x16x4_f32(false, A, false, B1, (short)0, Czero, false, false);
    v8f D2 = __builtin_amdgcn_wmma_f32_16x16x4_f32(false, A, false, B2, (short)0, Czero, false, false);

    // Lane L owns point base+L; fetch rows 0..6 of its column via bpermute.
    float c0,c1,c3,c4,c5,c6;
    {
        float x, y;
        x = __shfl(D1[0], laneMod16, 32); y = __shfl(D2[0], laneMod16, 32); c0 = hi ? y : x;
        x = __shfl(D1[1], laneMod16, 32); y = __shfl(D2[1], laneMod16, 32); c1 = hi ? y : x;
        x = __shfl(D1[3], laneMod16, 32); y = __shfl(D2[3], laneMod16, 32); c3 = hi ? y : x;
        x = __shfl(D1[4], laneMod16, 32); y = __shfl(D2[4], laneMod16, 32); c4 = hi ? y : x;
        x = __shfl(D1[5], laneMod16, 32); y = __shfl(D2[5], laneMod16, 32); c5 = hi ? y : x;
        x = __shfl(D1[6], laneMod16, 32); y = __shfl(D2[6], laneMod16, 32); c6 = hi ? y : x;
    }

    int  pp  = base + lane;
    bool inb = true;
    int  pc  = pp;
    if (TAIL) { inb = pp < N; pc = inb ? pp : (N - 1); }

    if (!TAIL) {
        // prefetch next grid-stride chunk (global_prefetch_b8)
        __builtin_prefetch(rotations + 4 * (pc + stride), 0, 1);
        __builtin_prefetch(scales    + 3 * (pc + stride), 0, 1);
        __builtin_prefetch(means3D   + 3 * (pc + stride), 0, 1);
    }

    // ---- per-point VALU math ----
    const float4 q = *(const float4*)(rotations + 4 * pc);   // (r,x,y,z)
    const float s0 = C.scale_mod * scales[3 * pc + 0];
    const float s1 = C.scale_mod * scales[3 * pc + 1];
    const float s2 = C.scale_mod * scales[3 * pc + 2];

    // projection results from WMMA (native rcp: 1-ulp, single v_rcp_f32)
    const float pwin = __builtin_amdgcn_rcpf(c3 + 1e-7f);
    const float ppx  = c0 * pwin;
    const float ppy  = c1 * pwin;
    const float pvx  = c4, pvy = c5, tz = c6;

    const float qr = q.x, qx = q.y, qy = q.z, qz = q.w;
    const float R00 = 1.f - 2.f*(qy*qy + qz*qz);
    const float R01 = 2.f*(qx*qy - qr*qz);
    const float R02 = 2.f*(qx*qz + qr*qy);
    const float R10 = 2.f*(qx*qy + qr*qz);
    const float R11 = 1.f - 2.f*(qx*qx + qz*qz);
    const float R12 = 2.f*(qy*qz - qr*qx);
    const float R20 = 2.f*(qx*qz - qr*qy);
    const float R21 = 2.f*(qy*qz + qr*qx);
    const float R22 = 1.f - 2.f*(qx*qx + qy*qy);

    const float w0 = s0*s0, w1 = s1*s1, w2 = s2*s2;
    const float S00 = w0*R00*R00 + w1*R10*R10 + w2*R20*R20;
    const float S01 = w0*R00*R01 + w1*R10*R11 + w2*R20*R21;
    const float S02 = w0*R00*R02 + w1*R10*R12 + w2*R20*R22;
    const float S11 = w0*R01*R01 + w1*R11*R11 + w2*R21*R21;
    const float S12 = w0*R01*R02 + w1*R11*R12 + w2*R21*R22;
    const float S22 = w0*R02*R02 + w1*R12*R12 + w2*R22*R22;

    const float invz  = __builtin_amdgcn_rcpf(tz);
    const float invz2 = invz * invz;
    const float txc = fminf(fmaxf(pvx * invz, -C.limx), C.limx) * tz;
    const float tyc = fminf(fmaxf(pvy * invz, -C.limy), C.limy) * tz;
    const float J00 =  C.focal_x * invz;
    const float J20 = -(C.focal_x * txc) * invz2;
    const float J11 =  C.focal_y * invz;
    const float J21 = -(C.focal_y * tyc) * invz2;
    const float u0 = C.vm0*J00 + C.vm2 *J20;
    const float u1 = C.vm4*J00 + C.vm6 *J20;
    const float u2 = C.vm8*J00 + C.vm10*J20;
    const float v0 = C.vm1*J11 + C.vm2 *J21;
    const float v1 = C.vm5*J11 + C.vm6 *J21;
    const float v2 = C.vm9*J11 + C.vm10*J21;

    const float Su0 = S00*u0 + S01*u1 + S02*u2;
    const float Su1 = S01*u0 + S11*u1 + S12*u2;
    const float Su2 = S02*u0 + S12*u1 + S22*u2;
    const float Sv0 = S00*v0 + S01*v1 + S02*v2;
    const float Sv1 = S01*v0 + S11*v1 + S12*v2;
    const float Sv2 = S02*v0 + S12*v1 + S22*v2;
    const float cov_x = u0*Su0 + u1*Su1 + u2*Su2 + 0.3f;
    const float cov_y = u0*Sv0 + u1*Sv1 + u2*Sv2;
    const float cov_z = v0*Sv0 + v1*Sv1 + v2*Sv2 + 0.3f;

    const float det = cov_x*cov_z - cov_y*cov_y;
    const float mid = 0.5f*(cov_x + cov_z);
    const float lam = mid + __builtin_amdgcn_sqrtf(fmaxf(0.1f, mid*mid - det));
    const float radius = __builtin_ceilf(3.0f * __builtin_amdgcn_sqrtf(lam));

    const float px = ((ppx + 1.0f)*C.W - 1.0f)*0.5f;
    const float py = ((ppy + 1.0f)*C.H - 1.0f)*0.5f;
    const float rminx = fminf(fmaxf(__builtin_floorf((px - radius)              * (1.0f/TILE_PX)), 0.0f), C.gx);
    const float rminy = fminf(fmaxf(__builtin_floorf((py - radius)              * (1.0f/TILE_PX)), 0.0f), C.gy);
    const float rmaxx = fminf(fmaxf(__builtin_floorf((px + radius + TILE_PX-1.f)* (1.0f/TILE_PX)), 0.0f), C.gx);
    const float rmaxy = fminf(fmaxf(__builtin_floorf((py + radius + TILE_PX-1.f)* (1.0f/TILE_PX)), 0.0f), C.gy);
    const float area  = (rmaxx - rminx) * (rmaxy - rminy);

    const bool valid = (tz > 0.2f) && (det != 0.0f) && (area != 0.0f);
    const float outr = valid ? radius : 0.0f;

    if (TAIL) { if (inb) radii[pp] = outr; }
    else      { radii[pp] = outr; }
}

__global__ __launch_bounds__(256) void gs_preprocess_kernel(
    const float* __restrict__ means3D,
    const float* __restrict__ scales,
    const float* __restrict__ rotations,
    const float* __restrict__ viewm,
    const float* __restrict__ projm,
    const float* __restrict__ p_tanfovx,
    const float* __restrict__ p_tanfovy,
    const float* __restrict__ p_scale_mod,
    const int*   __restrict__ p_imh,
    const int*   __restrict__ p_imw,
    float* __restrict__ radii,
    int N)
{
    const int  lane      = threadIdx.x & 31;
    const int  laneMod16 = lane & 15;
    const bool hi        = lane >= 16;
    const int  meanOff   = hi ? 2 : 0;

    const float tanfovx = p_tanfovx[0];
    const float tanfovy = p_tanfovy[0];
    GSCtx C;
    C.scale_mod = p_scale_mod[0];
    C.H = (float)p_imh[0];
    C.W = (float)p_imw[0];
    C.focal_x = C.W / (2.0f * tanfovx);
    C.focal_y = C.H / (2.0f * tanfovy);
    C.limx = 1.3f * tanfovx;
    C.limy = 1.3f * tanfovy;
    C.gx = (C.W + TILE_PX - 1.0f) / TILE_PX;
    C.gy = (C.H + TILE_PX - 1.0f) / TILE_PX;

    float vm[16], pm[16];
#pragma unroll
    for (int i = 0; i < 16; ++i) { vm[i] = viewm[i]; pm[i] = projm[i]; }
    C.vm0 = vm[0];  C.vm1 = vm[1];  C.vm2  = vm[2];
    C.vm4 = vm[4];  C.vm5 = vm[5];  C.vm6  = vm[6];
    C.vm8 = vm[8];  C.vm9 = vm[9];  C.vm10 = vm[10];

    // ---- A-matrix (16x4 f32): rows 0..3 = proj columns, rows 4..6 = view columns.
    // lanes 0-15 hold m=lane with (K0,K1); lanes 16-31 hold (K2,K3).
    const int m  = laneMod16;
    const int k0 = hi ? 2 : 0;
    float a0 = 0.0f, a1 = 0.0f;
    if (m < 4)      { a0 = pm[4*k0 + m];       a1 = pm[4*(k0+1) + m]; }
    else if (m < 7) { a0 = vm[4*k0 + (m - 4)]; a1 = vm[4*(k0+1) + (m - 4)]; }
    v2f A; A.x = a0; A.y = a1;

    const int waveGlobal = (int)((blockIdx.x * blockDim.x + threadIdx.x) >> 5);
    const int stride     = (int)(gridDim.x * blockDim.x);   // points per grid pass

    int base = waveGlobal * 32;
    // Main loop: whole 32-point chunks, no clamps, unconditional store.
    for (; base + 32 <= N; base += stride)
        gs_process32<false>(base, N, stride, lane, laneMod16, hi, meanOff, A, C,
                            means3D, scales, rotations, radii);
    // Tail: clamped loads, predicated store (EXEC still full at the WMMAs).
    if (base < N)
        gs_process32<true>(base, N, stride, lane, laneMod16, hi, meanOff, A, C,
                           means3D, scales, rotations, radii);
}

extern "C" void kernel_launch(void* const* d_in, const int* in_sizes, int n_in,
                              void* d_out, int out_size, void* d_ws, size_t ws_size,
                              hipStream_t stream) {
    const float* means3D   = (const float*)d_in[0];
    const float* scales    = (const float*)d_in[1];
    const float* rotations = (const float*)d_in[2];
    const float* viewm     = (const float*)d_in[3];
    const float* projm     = (const float*)d_in[4];
    const float* tanfovx   = (const float*)d_in[5];
    const float* tanfovy   = (const float*)d_in[6];
    const float* scale_mod = (const float*)d_in[7];
    const int*   imh       = (const int*)d_in[8];
    const int*   imw       = (const int*)d_in[9];
    float* radii = (float*)d_out;

    const int N = in_sizes[0] / 3;           // 4,000,000
    const int threads = 256;                 // 8 waves / block
    const int blocks  = 2048;                // grid-stride saturates HBM

    gs_preprocess_kernel<<<blocks, threads, 0, stream>>>(
        means3D, scales, rotations, viewm, projm,
        tanfovx, tanfovy, scale_mod, imh, imw, radii, N);
}